// MinkowskiUnion_18631568130260
// MI455X (gfx1250) — compile-verified
//
#include <hip/hip_runtime.h>

// MinkowskiUnion scatter-add for MI455X (gfx1250, wave32).
//
//   out = zeros(M, 64)
//   out[map1_out] += feat1[map1_in]
//   out[map2_out] += feat2[map2_in]
//
// Memory-bound: ~1.3-1.7 GB of HBM traffic -> ~60-75 us at 23.3 TB/s.
// No matmul structure => WMMA not applicable; the optimal gfx1250 path is
// native global_atomic_add_f32 (L2 atomic units) + nontemporal streamed
// feature reads + b128 zero fill.

typedef float v4f __attribute__((ext_vector_type(4)));
typedef float v2f __attribute__((ext_vector_type(2)));

#define C_DIM 64   // channels (fixed by the problem)
#define BLOCK 256  // 8 wave32 per block

// ---------------------------------------------------------------------------
// Zero-fill the output with 128-bit stores. Regular-temporal (RT) stores on
// purpose: the zeroed lines that stay resident in the 192MB L2 turn a chunk
// of the later atomic RMWs into L2 hits instead of HBM read-modify-writes.
// ---------------------------------------------------------------------------
__global__ void mu_zero_kernel(float* __restrict__ out, long long n4) {
  long long i = (long long)blockIdx.x * blockDim.x + threadIdx.x;
  if (i < n4) {
    v4f z = {0.f, 0.f, 0.f, 0.f};
    ((v4f*)out)[i] = z;  // global_store_b128
  }
}

// ---------------------------------------------------------------------------
// One wave32 per input row.
//   lane L owns channels [2L, 2L+1]  ->  one coalesced 256B row read per wave
// Gather with nontemporal hint (features are streamed exactly once; don't
// pollute L2, which we want full of output lines). Scatter with native
// fp32 global atomics at DEVICE scope (different workgroups may hit the
// same output row, so WGP-default scope would be incorrect).
// ---------------------------------------------------------------------------
__global__ void mu_scatter_add_kernel(const float* __restrict__ feat,
                                      const int*   __restrict__ map_in,
                                      const int*   __restrict__ map_out,
                                      float*       __restrict__ out,
                                      int n_rows) {
  const int lane = threadIdx.x & 31;
  const int row  = blockIdx.x * (BLOCK >> 5) + (threadIdx.x >> 5);
  if (row >= n_rows) return;

  const int in_row  = map_in[row];   // uniform per wave -> broadcast load
  const int out_row = map_out[row];

  // Streamed gather: 2 floats per lane, th:TH_LOAD_NT
  const v2f* src = (const v2f*)(feat + (long long)in_row * C_DIM);
  v2f v = __builtin_nontemporal_load(src + lane);

  // Device-scope no-return fp32 atomic adds (tracked by STOREcnt; the
  // implicit wait-idle at s_endpgm guarantees completion before the next
  // kernel / host validation).
  float* dst = out + (long long)out_row * C_DIM + 2 * lane;
  asm volatile("global_atomic_add_f32 %0, %1, off scope:SCOPE_DEV"
               :: "v"(dst),     "v"(v.x) : "memory");
  asm volatile("global_atomic_add_f32 %0, %1, off scope:SCOPE_DEV"
               :: "v"(dst + 1), "v"(v.y) : "memory");
}

// ---------------------------------------------------------------------------
// Host launcher. Input order (from setup_inputs):
//   d_in[0] feat1    (N1*64 f32)
//   d_in[1] feat2    (N2*64 f32)
//   d_in[2] map1_in  (N1 i32)
//   d_in[3] map1_out (N1 i32)
//   d_in[4] map2_in  (N2 i32)
//   d_in[5] map2_out (N2 i32)
//   d_in[6] out_size (scalar)
// d_out: M*64 f32 (out_size flat elements)
// ---------------------------------------------------------------------------
extern "C" void kernel_launch(void* const* d_in, const int* in_sizes, int n_in,
                              void* d_out, int out_size, void* d_ws, size_t ws_size,
                              hipStream_t stream) {
  const float* feat1    = (const float*)d_in[0];
  const float* feat2    = (const float*)d_in[1];
  const int*   map1_in  = (const int*)d_in[2];
  const int*   map1_out = (const int*)d_in[3];
  const int*   map2_in  = (const int*)d_in[4];
  const int*   map2_out = (const int*)d_in[5];
  float*       out      = (float*)d_out;

  const int n1 = in_sizes[2];  // rows of tensor 1
  const int n2 = in_sizes[4];  // rows of tensor 2

  // 1) zero-init output (out_size flat f32 elements, multiple of 4 since C=64)
  const long long n4 = (long long)out_size / 4;
  const int zgrid = (int)((n4 + BLOCK - 1) / BLOCK);
  mu_zero_kernel<<<zgrid, BLOCK, 0, stream>>>(out, n4);

  // 2) scatter-add both tensors; same stream => ordered after the zero pass
  const int rows_per_block = BLOCK / 32;
  const int g1 = (n1 + rows_per_block - 1) / rows_per_block;
  const int g2 = (n2 + rows_per_block - 1) / rows_per_block;
  mu_scatter_add_kernel<<<g1, BLOCK, 0, stream>>>(feat1, map1_in, map1_out, out, n1);
  mu_scatter_add_kernel<<<g2, BLOCK, 0, stream>>>(feat2, map2_in, map2_out, out, n2);
}